// GAGKNNQueryAndGroup_48215302865115
// MI455X (gfx1250) — compile-verified
//
#include <hip/hip_runtime.h>

typedef __attribute__((ext_vector_type(2))) float v2f;
typedef __attribute__((ext_vector_type(8))) float v8f;

#define B_      4
#define N_      8192
#define NPOINT_ 2048
#define C_      64
#define NS_     32
#define LAM_    0.5f
#define NWAVE_  4
#define NPW_    (N_ / NWAVE_)   // 2048 points scanned per wave

// Intra-wave LDS ordering fence: DS instructions from one wave are processed
// in-order by the LDS unit; this drains DScnt and stops compiler reordering,
// so cross-lane LDS communication within a wave needs no block barrier.
__device__ __forceinline__ void lds_fence() {
    asm volatile("s_wait_dscnt 0x0" ::: "memory");
}

// 4 waves per workgroup, one 16-query tile per workgroup. Each wave streams
// its private 2048-point range 32 points at a time: two independent
// V_WMMA_F32_16X16X4_F32 ops produce two 16x16 dot blocks (K=3 padded to 4).
// Distances are tested in registers against a mirrored kth-best threshold via
// a float min-reduction, and the whole insertion path is skipped with one
// wave ballot when nothing qualifies. Per-wave sorted top-32 lists are
// stable-merged at the end; all 4 waves then gather centered xyz + 64
// feature channels with non-temporal stores (output is write-once).
__global__ __launch_bounds__(32 * NWAVE_) void gag_knn_group_kernel(
    const float* __restrict__ xyz,      // (B, N, 3)
    const float* __restrict__ new_xyz,  // (B, NPOINT, 3)
    const int*   __restrict__ comp,     // (B, N)
    const int*   __restrict__ newcomp,  // (B, NPOINT)
    const float* __restrict__ feat,     // (B, C, N)
    float* __restrict__ out)            // (B, 3+C, NPOINT, NS)
{
    __shared__ float qx[16], qy[16], qz[16], qn[16];
    __shared__ int   qc[16];
    __shared__ float dtile[NWAVE_][16][34];   // tile A cols 0-15, tile B cols 17-32
    __shared__ float topd[NWAVE_][16][33];    // row pad 33; col 32 = merge sentinel
    __shared__ int   topi[NWAVE_][16][33];
    __shared__ int   mtop[16][NS_];           // merged indices

    const int wg   = blockIdx.x;
    const int b    = wg / (NPOINT_ / 16);
    const int p0   = (wg % (NPOINT_ / 16)) * 16;
    const int tid  = threadIdx.x;
    const int wv   = tid >> 5;
    const int lane = tid & 31;
    const int hi   = lane >> 4;
    const int lo   = lane & 15;

    constexpr bool SYMM = (LAM_ == 0.5f);     // both mask branches scale by 0.5

    // ---- load 16 queries, precompute |q|^2 and components ----
    if (tid < 16) {
        const float* q = new_xyz + ((size_t)b * NPOINT_ + p0 + tid) * 3;
        float x = q[0], y = q[1], z = q[2];
        qx[tid] = x; qy[tid] = y; qz[tid] = z;
        qn[tid] = x * x + y * y + z * z;
        qc[tid] = newcomp[(size_t)b * NPOINT_ + p0 + tid];
    }
    // init all top lists (incl. padded sentinel column 32 used by the merge)
    for (int j = tid; j < NWAVE_ * 16 * 33; j += 32 * NWAVE_) {
        int w = j / (16 * 33), rem = j % (16 * 33);
        topd[w][rem / 33][rem % 33] = 3.4e38f;
        topi[w][rem / 33][rem % 33] = 0;
    }
    __syncthreads();

    // A operand (16x4 f32): lane l -> row M=l%16; lo half K={0,1}, hi half K={2,3}
    v2f a;
    a.x = hi ? qz[lo] : qx[lo];
    a.y = hi ? 0.0f   : qy[lo];
    float qh8[8]; float qn8[8]; int nc8[8]; float kworst[8];
#pragma unroll
    for (int r = 0; r < 8; ++r) {
        qn8[r]    = qn[r + 8 * hi];
        qh8[r]    = 0.5f * qn8[r];            // pre-folded lambda scale (SYMM path)
        nc8[r]    = qc[r + 8 * hi];
        kworst[r] = 3.4e38f;
    }

    const float* xb = xyz  + (size_t)b * N_ * 3;
    const int*   cb = comp + (size_t)b * N_;

    const int nbeg = wv * NPW_;
    const int nend = nbeg + NPW_;
    for (int n0 = nbeg; n0 < nend; n0 += 32) {
        // speculative prefetch (OOB translation failures are silently dropped)
        __builtin_prefetch(xb + (size_t)(n0 + 32) * 3, 0, 1);

        const int nA = n0 + lo;               // both half-waves load point n = lo
        const int nB = n0 + 16 + lo;
        float pxA = xb[nA * 3 + 0], pyA = xb[nA * 3 + 1], pzA = xb[nA * 3 + 2];
        float pxB = xb[nB * 3 + 0], pyB = xb[nB * 3 + 1], pzB = xb[nB * 3 + 2];
        float phA = 0.5f * (pxA * pxA + pyA * pyA + pzA * pzA);
        float phB = 0.5f * (pxB * pxB + pyB * pyB + pzB * pzB);
        int   cnA = cb[nA], cnB = cb[nB];

        // B operands (4x16 f32): lo half K={0,1}, hi half K={2,3}
        v2f bA, bB;
        bA.x = hi ? pzA : pxA;  bA.y = hi ? 0.0f : pyA;
        bB.x = hi ? pzB : pxB;  bB.y = hi ? 0.0f : pyB;

        v8f cz   = {0.f, 0.f, 0.f, 0.f, 0.f, 0.f, 0.f, 0.f};
        v8f accA = __builtin_amdgcn_wmma_f32_16x16x4_f32(
            false, a, false, bA, (short)0, cz, false, false);
        v8f accB = __builtin_amdgcn_wmma_f32_16x16x4_f32(
            false, a, false, bB, (short)0, cz, false, false);

        // distances + register-resident threshold test as a float min-reduction
        float dA[8], dB[8];
        float hit = 3.4e38f;
#pragma unroll
        for (int r = 0; r < 8; ++r) {
            if (SYMM) {
                dA[r] = (qh8[r] + phA) - accA[r];   // 0.5*(|q|^2+|p|^2-2q.p)
                dB[r] = (qh8[r] + phB) - accB[r];
            } else {
                float sA = (cnA == nc8[r]) ? LAM_ : (1.0f - LAM_);
                float sB = (cnB == nc8[r]) ? LAM_ : (1.0f - LAM_);
                dA[r] = sA * (qn8[r] + 2.0f * phA - 2.0f * accA[r]);
                dB[r] = sB * (qn8[r] + 2.0f * phB - 2.0f * accB[r]);
            }
            hit = fminf(hit, dA[r] - kworst[r]);
            hit = fminf(hit, dB[r] - kworst[r]);
        }

        // wave-uniform skip: insertion path only when some lane has a candidate
        if (__builtin_amdgcn_ballot_w32(hit < 0.0f) != 0u) {
#pragma unroll
            for (int r = 0; r < 8; ++r) {
                dtile[wv][r + 8 * hi][lo]      = dA[r];
                dtile[wv][r + 8 * hi][17 + lo] = dB[r];
            }
            lds_fence();

            if (lane < 16) {                  // lane m owns query m, stable insert
                const int m = lane;
                float worst = topd[wv][m][NS_ - 1];
                for (int t = 0; t < 32; ++t) {    // ascending candidate index
                    const int col = (t < 16) ? t : (t + 1);
                    float d = dtile[wv][m][col];
                    if (d < worst) {          // strict: ties keep earlier index
                        int j = NS_ - 1;
                        while (j > 0 && topd[wv][m][j - 1] > d) {  // stop at == -> stable
                            topd[wv][m][j] = topd[wv][m][j - 1];
                            topi[wv][m][j] = topi[wv][m][j - 1];
                            --j;
                        }
                        topd[wv][m][j] = d;
                        topi[wv][m][j] = n0 + t;
                        worst = topd[wv][m][NS_ - 1];
                    }
                }
            }
            lds_fence();
#pragma unroll
            for (int r = 0; r < 8; ++r)       // refresh register thresholds
                kworst[r] = topd[wv][r + 8 * hi][NS_ - 1];
        }
    }
    __syncthreads();

    // ---- stable 4-way merge (index ranges ordered by wave; first-min wins ties)
    if (tid < 16) {
        const int m = tid;
        int p[NWAVE_] = {0, 0, 0, 0};
        for (int j = 0; j < NS_; ++j) {
            float best = topd[0][m][p[0]]; int bw = 0;
#pragma unroll
            for (int w = 1; w < NWAVE_; ++w) {
                float dw = topd[w][m][p[w]];
                if (dw < best) { best = dw; bw = w; }
            }
            mtop[m][j] = topi[bw][m][p[bw]];
            p[bw]++;
        }
    }
    __syncthreads();

    // ---- gather + write: wave wv handles 4 queries; lane s -> sample s ----
    const float* fb   = feat + (size_t)b * C_ * N_;
    const size_t outb = (size_t)b * (3 + C_) * NPOINT_ * NS_;
    const size_t chs  = (size_t)NPOINT_ * NS_;

    for (int mm = 0; mm < 16 / NWAVE_; ++mm) {
        const int m  = wv * (16 / NWAVE_) + mm;
        const int id = mtop[m][lane];
        float gx = xb[id * 3 + 0] - qx[m];
        float gy = xb[id * 3 + 1] - qy[m];
        float gz = xb[id * 3 + 2] - qz[m];
        size_t base = outb + (size_t)(p0 + m) * NS_ + lane;
        __builtin_nontemporal_store(gx, &out[base + 0 * chs]);
        __builtin_nontemporal_store(gy, &out[base + 1 * chs]);
        __builtin_nontemporal_store(gz, &out[base + 2 * chs]);
#pragma unroll 4
        for (int c = 0; c < C_; ++c) {
            __builtin_nontemporal_store(fb[(size_t)c * N_ + id],
                                        &out[base + (size_t)(3 + c) * chs]);
        }
    }
}

extern "C" void kernel_launch(void* const* d_in, const int* in_sizes, int n_in,
                              void* d_out, int out_size, void* d_ws, size_t ws_size,
                              hipStream_t stream) {
    (void)in_sizes; (void)n_in; (void)out_size; (void)d_ws; (void)ws_size;
    const float* xyz     = (const float*)d_in[0];
    const float* new_xyz = (const float*)d_in[1];
    const int*   comp    = (const int*)  d_in[2];
    const int*   newcomp = (const int*)  d_in[3];
    const float* feat    = (const float*)d_in[4];
    float*       out     = (float*)d_out;

    dim3 grid(B_ * NPOINT_ / 16);    // 512 workgroups x 4 waves = 2048 waves
    dim3 block(32 * NWAVE_);
    gag_knn_group_kernel<<<grid, block, 0, stream>>>(xyz, new_xyz, comp, newcomp, feat, out);
}